// SphericalHarmonicTransform_36369783062807
// MI455X (gfx1250) — compile-verified
//
#include <hip/hip_runtime.h>
#include <hip/hip_bf16.h>

typedef __attribute__((ext_vector_type(2))) float v2f;
typedef __attribute__((ext_vector_type(8))) float v8f;

#define NLAT   256
#define NBC    512      // 4*128 batch rows
#define TOTAL  70144    // sum(NLON)
#define MMAXC  256
#define LMAXC  256
#define MAXL   528
#define APITCH 260      // LDS row pitch (floats): bank stride 4/row -> conflict-free
#define TWO_PI 6.28318530717958647692f

// ring i < 128:  L = 20+4i,   offset = 2i^2 + 18i
// ring i >= 128: j=i-128, L = 528-4j, offset = 35072 + 528j - 2j(j-1)
__device__ __forceinline__ void ring_geom(int ring, int* L, int* off) {
    if (ring < 128) {
        *L   = 20 + 4 * ring;
        *off = 2 * ring * ring + 18 * ring;
    } else {
        int j = ring - 128;
        *L   = 528 - 4 * j;
        *off = 35072 + 528 * j - 2 * j * (j - 1);
    }
}

// ---------------------------------------------------------------------------
// Stage 1: per-ring real DFT.  One workgroup = (one ring) x (8 batch rows).
// cos/sin table of 2*pi*j/L built once in LDS; twiddle index (m*n) mod L is
// maintained incrementally -> zero transcendentals in the inner loop.
// Output layout Xr/Xi: [m][k][bc]  (bc contiguous -> two b128 stores/thread).
// Writes zeros for m > L/2 so stage 2 never reads poisoned workspace.
// ---------------------------------------------------------------------------
__global__ __launch_bounds__(256) void sht_dft_kernel(const float* __restrict__ x,
                                                      float* __restrict__ Xr,
                                                      float* __restrict__ Xi) {
    __shared__ float costab[MAXL];
    __shared__ float sintab[MAXL];
    __shared__ float xs[8][MAXL];

    const int ring = blockIdx.x >> 6;
    const int bc0  = (blockIdx.x & 63) * 8;
    int L, off;
    ring_geom(ring, &L, &off);

    const int   tid  = threadIdx.x;
    const float invL = 1.0f / (float)L;

    for (int j = tid; j < L; j += 256) {
        float s, c;
        __sincosf(TWO_PI * (float)j * invL, &s, &c);
        costab[j] = c;
        sintab[j] = s;
    }
    for (int idx = tid; idx < 8 * L; idx += 256) {
        int r = idx / L;
        int n = idx - r * L;
        xs[r][n] = x[(size_t)(bc0 + r) * TOTAL + off + n];
    }
    __syncthreads();

    const int m = tid;   // one output frequency per thread
    float accR[8] = {0.f,0.f,0.f,0.f,0.f,0.f,0.f,0.f};
    float accI[8] = {0.f,0.f,0.f,0.f,0.f,0.f,0.f,0.f};

    if (m <= (L >> 1)) {
        int phase = 0;
        for (int n = 0; n < L; ++n) {
            float c = costab[phase];
            float s = sintab[phase];
            #pragma unroll
            for (int r = 0; r < 8; ++r) {
                float xv = xs[r][n];
                accR[r] = __builtin_fmaf(xv,  c, accR[r]);
                accI[r] = __builtin_fmaf(xv, -s, accI[r]);
            }
            phase += m;
            if (phase >= L) phase -= L;     // (m*n) mod L, incrementally
        }
        const float scale = TWO_PI * invL;  // 2*pi * forward-norm(1/L)
        #pragma unroll
        for (int r = 0; r < 8; ++r) { accR[r] *= scale; accI[r] *= scale; }
    }

    // [m][k][bc]: 8 consecutive bc floats -> two b128 stores per array
    const size_t base = (size_t)m * (NLAT * NBC) + (size_t)ring * NBC + (size_t)bc0;
    *(float4*)(Xr + base)     = make_float4(accR[0], accR[1], accR[2], accR[3]);
    *(float4*)(Xr + base + 4) = make_float4(accR[4], accR[5], accR[6], accR[7]);
    *(float4*)(Xi + base)     = make_float4(accI[0], accI[1], accI[2], accI[3]);
    *(float4*)(Xi + base + 4) = make_float4(accI[4], accI[5], accI[6], accI[7]);
}

// ---------------------------------------------------------------------------
// Stage 2: for each m, GEMM  C[bc,l] = sum_k A[bc,k] * W[m,l,k]  via
// V_WMMA_F32_16X16X4_F32.  One block = (one m, one 16-row bc tile):
//  - A re/im panels (16 x 256) staged in LDS once (transposed from [k][bc]),
//    padded pitch 260 -> WMMA fragment ds_load_b64 hits all 64 banks.
//  - 8 waves x 2 l-tiles each cover l = 0..255; each wave holds 4
//    accumulators (re/im x 2 l-tiles) sharing the LDS A fragments.
//  - weight stream prefetched one 64-k chunk ahead (global_prefetch_b8).
// Output stored as interleaved complex64 with one b64 store per element.
// ---------------------------------------------------------------------------
__global__ __launch_bounds__(256) void sht_leg_wmma_kernel(const float* __restrict__ Xr,
                                                           const float* __restrict__ Xi,
                                                           const float* __restrict__ W,
                                                           float* __restrict__ out) {
    __shared__ float ArL[16 * APITCH];
    __shared__ float AiL[16 * APITCH];

    const int tid  = threadIdx.x;
    const int m    = blockIdx.x >> 5;          // blocks with same m are adjacent
    const int bc0  = (blockIdx.x & 31) * 16;

    // ---- cooperative stage of A panels: global [m][k][bc] -> LDS [bc][k] ----
    {
        const int k = tid;                      // one k row per thread
        const float4* gr = (const float4*)(Xr + (size_t)m * (NLAT * NBC)
                                              + (size_t)k * NBC + bc0);
        const float4* gi = (const float4*)(Xi + (size_t)m * (NLAT * NBC)
                                              + (size_t)k * NBC + bc0);
        float vr[16], vi[16];
        #pragma unroll
        for (int q = 0; q < 4; ++q) {
            *(float4*)(vr + 4 * q) = gr[q];
            *(float4*)(vi + 4 * q) = gi[q];
        }
        #pragma unroll
        for (int j = 0; j < 16; ++j) {          // transpose scatter into LDS
            ArL[j * APITCH + k] = vr[j];
            AiL[j * APITCH + k] = vi[j];
        }
    }
    __syncthreads();

    const int lane = tid & 31;
    const int wave = tid >> 5;
    const int l0   = wave << 5;                 // wave owns l0..l0+31 (2 l-tiles)
    const int row  = lane & 15;                 // A: M row / B,C: N col
    const int grp  = lane >> 4;                 // k sub-offset (k+0,1 vs k+2,3)

    const float* aR = ArL + row * APITCH + grp * 2;
    const float* aI = AiL + row * APITCH + grp * 2;
    const float* b0 = W + (size_t)m * (LMAXC * NLAT) + (size_t)(l0 + row) * NLAT + grp * 2;
    const float* b1 = b0 + 16 * NLAT;

    v8f cR0 = {}, cI0 = {}, cR1 = {}, cI1 = {};

    for (int k0 = 0; k0 < NLAT; k0 += 64) {
        if (k0 + 64 < NLAT) {                   // prefetch next weight chunk
            __builtin_prefetch(b0 + k0 + 64, 0, 1);
            __builtin_prefetch(b1 + k0 + 64, 0, 1);
        }
        #pragma unroll 4
        for (int kk = 0; kk < 64; kk += 4) {
            const int k = k0 + kk;
            v2f fr = *(const v2f*)(aR + k);     // ds_load_b64
            v2f fi = *(const v2f*)(aI + k);
            v2f w0 = *(const v2f*)(b0 + k);     // global_load_b64
            v2f w1 = *(const v2f*)(b1 + k);
            cR0 = __builtin_amdgcn_wmma_f32_16x16x4_f32(false, fr, false, w0, (short)0,
                                                        cR0, false, false);
            cI0 = __builtin_amdgcn_wmma_f32_16x16x4_f32(false, fi, false, w0, (short)0,
                                                        cI0, false, false);
            cR1 = __builtin_amdgcn_wmma_f32_16x16x4_f32(false, fr, false, w1, (short)0,
                                                        cR1, false, false);
            cI1 = __builtin_amdgcn_wmma_f32_16x16x4_f32(false, fi, false, w1, (short)0,
                                                        cI1, false, false);
        }
    }

    // C layout: VGPR r -> M = r (lanes 0-15) / r+8 (lanes 16-31); N = lane&15
    float2* out2 = (float2*)out;                // complex64 elements
    const size_t ob = (size_t)(bc0 + grp * 8) * (LMAXC * MMAXC)
                    + (size_t)(l0 + row) * MMAXC + (size_t)m;
    #pragma unroll
    for (int r = 0; r < 8; ++r) {
        out2[ob + (size_t)r * (LMAXC * MMAXC)]              = make_float2(cR0[r], cI0[r]);
        out2[ob + (size_t)r * (LMAXC * MMAXC) + 16 * MMAXC] = make_float2(cR1[r], cI1[r]);
    }
}

extern "C" void kernel_launch(void* const* d_in, const int* in_sizes, int n_in,
                              void* d_out, int out_size, void* d_ws, size_t ws_size,
                              hipStream_t stream) {
    const float* x = (const float*)d_in[0];   // (4,128,70144) f32
    const float* w = (const float*)d_in[1];   // (256,256,256) f32

    float* Xr = (float*)d_ws;                              // [m][k][bc]
    float* Xi = Xr + (size_t)MMAXC * NLAT * NBC;           // +33.5M floats

    // Stage 1: 256 rings x 64 bc-tiles
    sht_dft_kernel<<<256 * 64, 256, 0, stream>>>(x, Xr, Xi);

    // Stage 2: 256 m x 32 bc-tiles, 8 waves x 2 l-tiles per block
    sht_leg_wmma_kernel<<<256 * 32, 256, 0, stream>>>(Xr, Xi, w, (float*)d_out);
}